// SimpleGCN_43173011259900
// MI455X (gfx1250) — compile-verified
//
#include <hip/hip_runtime.h>
#include <hip/hip_bf16.h>

// ---------------------------------------------------------------------------
// 2-layer GCN (DGL GraphConv norm='both') for MI455X / gfx1250.
//   h = relu( Din^-1/2 * A * Dout^-1/2 * X * W1 + b1 )
//   y =       Din^-1/2 * A * Dout^-1/2 * h * W2 + b2
// Memory-bound workload (~1.4 GB of SpMM traffic vs <1.1 GFLOP of GEMM):
//   - SpMM: edge-parallel float4 gather + global_atomic_add_f32 scatter
//     (25.6 MB feature matrix lives in the 192 MB L2)
//   - GEMM: exact FP32 via V_WMMA_F32_16X16X4_F32 (wave32 WMMA), tiles staged
//     in LDS via gfx1250 async global->LDS copies (ASYNCcnt).
//   - Din^-1/2 row scaling commutes with the GEMM, applied to the accumulator
//     rows in the epilogue so tile staging is a pure DMA copy.
// ---------------------------------------------------------------------------

#define N_NODES   100000
#define IN_DIM    64
#define HID_DIM   64
#define N_CLASSES 16

typedef __attribute__((ext_vector_type(2))) float v2f;
typedef __attribute__((ext_vector_type(8))) float v8f;
typedef __attribute__((ext_vector_type(4))) int   v4i;

// ---------------- gfx1250 async global->LDS helpers ----------------

#if __has_builtin(__builtin_amdgcn_global_load_async_to_lds_b128)
#define GCN_HAS_ASYNC_LDS 1
#else
#define GCN_HAS_ASYNC_LDS 0
#endif

// Pointee types the async builtin expects: b128 = v4i, in AS1 (global) / AS3 (LDS).
typedef __attribute__((address_space(1))) v4i* gcn_gptr128;
typedef __attribute__((address_space(3))) v4i* gcn_lptr128;

// Copy 16 bytes global -> LDS. Async path writes LDS directly (no VGPR
// round-trip), tracked by ASYNCcnt.
__device__ __forceinline__ void lds_copy16(void* lds, const void* gmem) {
#if GCN_HAS_ASYNC_LDS
    __builtin_amdgcn_global_load_async_to_lds_b128(
        (gcn_gptr128)gmem, (gcn_lptr128)lds, /*imm offset=*/0, /*cpol=*/0);
#else
    *(float4*)lds = *(const float4*)gmem;
#endif
}

__device__ __forceinline__ void wait_lds_copies() {
#if GCN_HAS_ASYNC_LDS
#if __has_builtin(__builtin_amdgcn_s_wait_asynccnt)
    __builtin_amdgcn_s_wait_asynccnt(0);
#else
    asm volatile("s_wait_asynccnt 0" ::: "memory");
#endif
#endif
}

// ---------------- utility kernels ----------------

__global__ void gcn_zero_f32(float* __restrict__ p, int n) {
    int i = blockIdx.x * blockDim.x + threadIdx.x;
    if (i < n) p[i] = 0.0f;
}

__global__ void gcn_degrees(const int* __restrict__ src, const int* __restrict__ dst,
                            float* __restrict__ deg_out, float* __restrict__ deg_in,
                            int n_edges) {
    int e = blockIdx.x * blockDim.x + threadIdx.x;
    if (e >= n_edges) return;
    atomicAdd(&deg_out[src[e]], 1.0f);
    atomicAdd(&deg_in[dst[e]], 1.0f);
}

__global__ void gcn_rsqrt_inplace(float* __restrict__ p, int n) {
    int i = blockIdx.x * blockDim.x + threadIdx.x;
    if (i < n) {
        float d = p[i];
        p[i] = rsqrtf(d > 1.0f ? d : 1.0f);
    }
}

// ---------------- SpMM: agg[dst] += inv_out[src] * X[src] ----------------
// 16 lanes per edge, float4 per lane (64 features).

__global__ void gcn_spmm_scatter(const float* __restrict__ X,
                                 const int* __restrict__ src,
                                 const int* __restrict__ dst,
                                 const float* __restrict__ inv_out,
                                 float* __restrict__ agg,
                                 int n_edges) {
    long t = (long)blockIdx.x * blockDim.x + threadIdx.x;
    long e = t >> 4;
    if (e >= n_edges) return;
    int fg = (int)(t & 15) << 2;                 // feature group: 0,4,...,60
    int s = src[e], d = dst[e];
    float sc = inv_out[s];
    float4 v = *(const float4*)(X + (long)s * IN_DIM + fg);
    float* o = agg + (long)d * IN_DIM + fg;
    atomicAdd(o + 0, v.x * sc);
    atomicAdd(o + 1, v.y * sc);
    atomicAdd(o + 2, v.z * sc);
    atomicAdd(o + 3, v.w * sc);
}

// ---------------- GEMM1: H = relu(inv_in * (agg @ W1) + b1) ----------------
// M=100000 (6250 row-tiles of 16), N=64 (4 waves, one 16-col tile each),
// K=64 (16 x K=4 WMMA steps). W1 and the 16x64 A tile staged in LDS via
// async global->LDS copies.
// FP32 A layout (16x4): lane%16 = row M, lane/16 selects K pair {0,1}|{2,3}.
// FP32 C/D layout:      VGPR v -> M = v + 8*(lane/16), N = lane%16.

__global__ void __launch_bounds__(128)
gcn_gemm1_relu(const float* __restrict__ A,       // agg, N_NODES x 64
               const float* __restrict__ inv_in,
               const float* __restrict__ W,       // 64 x 64 (row-major K x N)
               const float* __restrict__ bias,    // 64
               float* __restrict__ H) {           // N_NODES x 64
    __shared__ float sW[64 * 64];
    __shared__ float sA[16 * 64];
    const int tid = threadIdx.x;
    const long row0 = (long)blockIdx.x * 16;

    // Stage W1 (1024 x b128) and 16x64 A tile (256 x b128): pure copies.
    const float4* gW = (const float4*)W;
    const float4* gA = (const float4*)(A + row0 * 64);
#pragma unroll
    for (int i = 0; i < 8; ++i)
        lds_copy16((float4*)sW + tid + 128 * i, gW + tid + 128 * i);
#pragma unroll
    for (int i = 0; i < 2; ++i)
        lds_copy16((float4*)sA + tid + 128 * i, gA + tid + 128 * i);
    wait_lds_copies();
    __syncthreads();

    const int lane  = tid & 31;
    const int wave  = tid >> 5;        // N tile: columns [wave*16, wave*16+16)
    const int m     = lane & 15;
    const int khalf = lane >> 4;       // 0 or 1
    const int n     = (wave << 4) + m; // this lane's B/D column

    v8f acc = {};
#pragma unroll
    for (int s = 0; s < 16; ++s) {
        const int kb = 4 * s + 2 * khalf;
        v2f a, b;
        a.x = sA[m * 64 + kb];
        a.y = sA[m * 64 + kb + 1];
        b.x = sW[kb * 64 + n];
        b.y = sW[(kb + 1) * 64 + n];
        acc = __builtin_amdgcn_wmma_f32_16x16x4_f32(
            /*neg_a=*/false, a, /*neg_b=*/false, b,
            /*c_mod=*/(short)0, acc, /*reuse_a=*/false, /*reuse_b=*/false);
    }

    // Epilogue: row-scale (Din^-1/2 commutes with GEMM), bias, ReLU.
    const float bcol = bias[n];
#pragma unroll
    for (int v = 0; v < 8; ++v) {
        const int mr = v + 8 * khalf;
        float val = acc[v] * inv_in[row0 + mr] + bcol;
        val = val > 0.0f ? val : 0.0f;
        H[(row0 + mr) * 64 + n] = val;
    }
}

// ---------------- GEMM2: OUT = inv_in * (agg2 @ W2) + b2 ----------------
// N=16 -> one 16x16 tile per wave; 8 waves/block, each a distinct row tile.

__global__ void __launch_bounds__(256)
gcn_gemm2(const float* __restrict__ A,        // agg2, N_NODES x 64
          const float* __restrict__ inv_in,
          const float* __restrict__ W,        // 64 x 16
          const float* __restrict__ bias,     // 16
          float* __restrict__ OUT,            // N_NODES x 16
          int n_rowtiles) {
    __shared__ float sW[64 * 16];
    const int tid = threadIdx.x;
    // Stage W2 (256 x b128), one copy per thread.
    lds_copy16((float4*)sW + tid, (const float4*)W + tid);
    wait_lds_copies();
    __syncthreads();

    const int wave    = tid >> 5;
    const int rowtile = blockIdx.x * 8 + wave;
    if (rowtile >= n_rowtiles) return;        // wave-uniform exit (EXEC stays full)

    const int lane  = tid & 31;
    const int m     = lane & 15;
    const int khalf = lane >> 4;
    const long row0 = (long)rowtile * 16;

    const float2* pA = (const float2*)(A + (row0 + m) * 64);

    v8f acc = {};
#pragma unroll
    for (int s = 0; s < 16; ++s) {
        const int kb = 4 * s + 2 * khalf;
        float2 av = pA[2 * s + khalf];        // A[m][kb], A[m][kb+1]
        v2f a, b;
        a.x = av.x;
        a.y = av.y;
        b.x = sW[kb * 16 + m];                // W2[kb][n], n == m for D column
        b.y = sW[(kb + 1) * 16 + m];
        acc = __builtin_amdgcn_wmma_f32_16x16x4_f32(
            false, a, false, b, (short)0, acc, false, false);
    }

    const float bcol = bias[m];
#pragma unroll
    for (int v = 0; v < 8; ++v) {
        const int mr = v + 8 * khalf;
        OUT[(row0 + mr) * 16 + m] = acc[v] * inv_in[row0 + mr] + bcol;
    }
}

// ---------------- host-side orchestration ----------------

extern "C" void kernel_launch(void* const* d_in, const int* in_sizes, int n_in,
                              void* d_out, int out_size, void* d_ws, size_t ws_size,
                              hipStream_t stream) {
    const float* features = (const float*)d_in[0];
    const int*   src      = (const int*)d_in[1];
    const int*   dst      = (const int*)d_in[2];
    const float* W1       = (const float*)d_in[3];
    const float* b1       = (const float*)d_in[4];
    const float* W2       = (const float*)d_in[5];
    const float* b2       = (const float*)d_in[6];
    float*       out      = (float*)d_out;

    const int n_edges = in_sizes[1];
    const int n_nodes = in_sizes[0] / IN_DIM;        // 100000

    // Workspace layout (floats): inv_out | inv_in | agg | h1  (~52.0 MB)
    float* inv_out = (float*)d_ws;
    float* inv_in  = inv_out + n_nodes;
    float* agg     = inv_in + n_nodes;
    float* h1      = agg + (size_t)n_nodes * HID_DIM;

    const int B = 256;
    const int n_deg  = 2 * n_nodes;                  // inv_out+inv_in contiguous
    const int n_agg  = n_nodes * HID_DIM;
    const long n_sc  = (long)n_edges * 16;           // scatter threads

    // degrees -> inv_sqrt (in place over both arrays)
    gcn_zero_f32<<<(n_deg + B - 1) / B, B, 0, stream>>>(inv_out, n_deg);
    gcn_degrees<<<(n_edges + B - 1) / B, B, 0, stream>>>(src, dst, inv_out, inv_in, n_edges);
    gcn_rsqrt_inplace<<<(n_deg + B - 1) / B, B, 0, stream>>>(inv_out, n_deg);

    // Layer 1: SpMM + WMMA GEMM (+bias, ReLU)
    gcn_zero_f32<<<(n_agg + B - 1) / B, B, 0, stream>>>(agg, n_agg);
    gcn_spmm_scatter<<<(int)((n_sc + B - 1) / B), B, 0, stream>>>(
        features, src, dst, inv_out, agg, n_edges);
    gcn_gemm1_relu<<<n_nodes / 16, 128, 0, stream>>>(agg, inv_in, W1, b1, h1);

    // Layer 2: SpMM (agg buffer reused) + WMMA GEMM (+bias)
    gcn_zero_f32<<<(n_agg + B - 1) / B, B, 0, stream>>>(agg, n_agg);
    gcn_spmm_scatter<<<(int)((n_sc + B - 1) / B), B, 0, stream>>>(
        h1, src, dst, inv_out, agg, n_edges);
    const int n_rowtiles = n_nodes / 16;             // 6250
    gcn_gemm2<<<(n_rowtiles + 7) / 8, 256, 0, stream>>>(agg, inv_in, W2, b2, out, n_rowtiles);
}